// RoiSelector_8358006358575
// MI455X (gfx1250) — compile-verified
//
#include <hip/hip_runtime.h>
#include <hip/hip_bf16.h>
#include <stdint.h>

#define BATCH   8
#define NPROP   200000
#define PPRE    2000
#define PPOST   1000
#define KWORDS  64        // 64 u32 words cover 2048 >= PPRE bits per mask row
#define NBUCKET 4096
#define CAP     4096
#define SORT_M  4096
#define NMS_TH  0.7f

// ---------------- workspace layout (bytes) ----------------
static constexpr size_t WS_CAND = 0;                                   // B*CAP u64
static constexpr size_t WS_HIST = WS_CAND + (size_t)BATCH * CAP * 8;   // B*NBUCKET u32
static constexpr size_t WS_THR  = WS_HIST + (size_t)BATCH * NBUCKET * 4; // B u32
static constexpr size_t WS_CCNT = WS_THR + (size_t)BATCH * 4;          // B u32
static constexpr size_t WS_TSC  = WS_CCNT + (size_t)BATCH * 4;         // B*PPRE f32
static constexpr size_t WS_TIDX = WS_TSC + (size_t)BATCH * PPRE * 4;   // B*PPRE i32
static constexpr size_t WS_TBOX = WS_TIDX + (size_t)BATCH * PPRE * 4;  // B*PPRE*4 f32
static constexpr size_t WS_KEEP = WS_TBOX + (size_t)BATCH * PPRE * 16; // B*PPRE u32
static constexpr size_t WS_MASK = WS_KEEP + (size_t)BATCH * PPRE * 4;  // B*PPRE*KWORDS u32

// order-preserving bijection: float -> u32 with ascending unsigned order
__device__ __forceinline__ uint32_t fkey(float f) {
  uint32_t u = __float_as_uint(f);
  uint32_t m = (uint32_t)((int32_t)u >> 31) | 0x80000000u;
  return u ^ m;
}

__global__ void k_init(uint32_t* hist, uint32_t* ccnt) {
  int i = blockIdx.x * blockDim.x + threadIdx.x;
  if (i < BATCH * NBUCKET) hist[i] = 0u;
  if (i < BATCH) ccnt[i] = 0u;
}

__global__ __launch_bounds__(256) void k_hist(const float* __restrict__ score,
                                              uint32_t* __restrict__ hist) {
  __shared__ uint32_t h[NBUCKET];
  int b = blockIdx.y;
  for (int i = threadIdx.x; i < NBUCKET; i += 256) h[i] = 0u;
  __syncthreads();
  int base = blockIdx.x * 2048;
  for (int k = 0; k < 8; ++k) {
    int i = base + k * 256 + threadIdx.x;
    if (i < NPROP) {
      uint32_t key = fkey(score[(size_t)b * NPROP + i]);
      atomicAdd(&h[key >> 20], 1u);
    }
  }
  __syncthreads();
  for (int i = threadIdx.x; i < NBUCKET; i += 256)
    if (h[i]) atomicAdd(&hist[b * NBUCKET + i], h[i]);
}

__global__ void k_thresh(const uint32_t* __restrict__ hist, uint32_t* __restrict__ thrT) {
  if (threadIdx.x == 0) {
    int b = blockIdx.x;
    uint32_t cum = 0; int T = 0;
    for (int bk = NBUCKET - 1; bk >= 0; --bk) {
      cum += hist[b * NBUCKET + bk];
      if (cum >= PPRE) { T = bk; break; }
    }
    thrT[b] = (uint32_t)T;
  }
}

__global__ __launch_bounds__(256) void k_compact(const float* __restrict__ score,
                                                 const uint32_t* __restrict__ thrT,
                                                 uint32_t* __restrict__ ccnt,
                                                 uint64_t* __restrict__ cand) {
  int b = blockIdx.y;
  uint32_t T = thrT[b];
  int base = blockIdx.x * 2048;
  for (int k = 0; k < 8; ++k) {
    int i = base + k * 256 + threadIdx.x;
    if (i < NPROP) {
      uint32_t key = fkey(score[(size_t)b * NPROP + i]);
      if ((key >> 20) >= T) {
        uint32_t pos = atomicAdd(&ccnt[b], 1u);
        if (pos < CAP)
          cand[(size_t)b * CAP + pos] = ((uint64_t)(~key) << 32) | (uint32_t)i;
      }
    }
  }
}

// one block per batch: bitonic sort 4096 keys in LDS, then decode top-2000
__global__ __launch_bounds__(1024) void k_sort_decode(const float* __restrict__ score,
                                                      const float* __restrict__ bbox,
                                                      const float* __restrict__ anchors,
                                                      const uint32_t* __restrict__ ccnt,
                                                      const uint64_t* __restrict__ cand,
                                                      float* __restrict__ tsc,
                                                      int* __restrict__ tidx,
                                                      float* __restrict__ tbox) {
  __shared__ uint64_t sk[SORT_M];
  int b = blockIdx.x;
  uint32_t nc = ccnt[b];
  int n = (nc > (uint32_t)CAP) ? CAP : (int)nc;
  for (int i = threadIdx.x; i < SORT_M; i += 1024)
    sk[i] = (i < n) ? cand[(size_t)b * CAP + i] : 0xFFFFFFFFFFFFFFFFull;
  __syncthreads();
  for (int k = 2; k <= SORT_M; k <<= 1) {
    for (int j = k >> 1; j > 0; j >>= 1) {
      for (int i = threadIdx.x; i < SORT_M; i += 1024) {
        int ixj = i ^ j;
        if (ixj > i) {
          uint64_t a = sk[i], c = sk[ixj];
          bool up = ((i & k) == 0);
          if ((a > c) == up) { sk[i] = c; sk[ixj] = a; }
        }
      }
      __syncthreads();
    }
  }
  for (int i = threadIdx.x; i < PPRE; i += 1024) {
    uint64_t kk = sk[i];
    uint32_t idx = (uint32_t)kk;
    if (idx >= NPROP) idx = 0;  // safety (cannot happen: N >> PPRE)
    float s = score[(size_t)b * NPROP + idx];
    float sg = 1.0f / (1.0f + expf(-s));
    const float* a = anchors + (size_t)idx * 4;
    const float* d = bbox + ((size_t)b * NPROP + idx) * 4;
    float aw = a[2] - a[0], ah = a[3] - a[1];
    float acx = a[0] + 0.5f * aw, acy = a[1] + 0.5f * ah;
    float cx = acx + d[0] * aw, cy = acy + d[1] * ah;
    float w = aw * expf(d[2]), h = ah * expf(d[3]);
    float* ob = tbox + ((size_t)b * PPRE + i) * 4;
    ob[0] = cx - 0.5f * w; ob[1] = cy - 0.5f * h;
    ob[2] = cx + 0.5f * w; ob[3] = cy + 0.5f * h;
    tsc[(size_t)b * PPRE + i] = sg;
    tidx[(size_t)b * PPRE + i] = (int)idx;
  }
}

// suppression bitmask: bit (r, c) set iff c>r && IoU(r,c) > 0.7
#define MROWS 128
__global__ __launch_bounds__(256) void k_mask(const float* __restrict__ tbox,
                                              uint32_t* __restrict__ maskM) {
  __shared__ float4 sbox[2048];
  int b = blockIdx.y;
  const float* tb = tbox + (size_t)b * PPRE * 4;

  // CDNA5 async-to-LDS staging of the 2000 column boxes (32 KB)
  uint32_t lbase = (uint32_t)(uintptr_t)(&sbox[0]);
  for (int i = threadIdx.x; i < PPRE; i += 256) {
    uint64_t g = (uint64_t)(uintptr_t)(tb + (size_t)i * 4);
    uint32_t l = lbase + (uint32_t)i * 16u;
    asm volatile("global_load_async_to_lds_b128 %0, %1, off"
                 :: "v"(l), "v"(g) : "memory");
  }
  asm volatile("s_wait_asynccnt 0x0" ::: "memory");
  __syncthreads();

  int r = blockIdx.x * MROWS + (threadIdx.x >> 1);
  if (r < PPRE) {
    float4 rb = *(const float4*)(tb + (size_t)r * 4);
    float rw = fmaxf(rb.z - rb.x, 0.0f), rh = fmaxf(rb.w - rb.y, 0.0f);
    float rarea = rw * rh;
    int w0 = (threadIdx.x & 1) * 32;
    uint32_t* out = maskM + ((size_t)b * PPRE + r) * KWORDS;
    for (int w = w0; w < w0 + 32; ++w) {
      uint32_t bits = 0u;
      int cbase = w * 32;
      for (int t = 0; t < 32; ++t) {
        int c = cbase + t;
        float4 cb = sbox[c & 2047];
        float cw = fmaxf(cb.z - cb.x, 0.0f), ch = fmaxf(cb.w - cb.y, 0.0f);
        float carea = cw * ch;
        float ix1 = fmaxf(rb.x, cb.x), iy1 = fmaxf(rb.y, cb.y);
        float ix2 = fminf(rb.z, cb.z), iy2 = fminf(rb.w, cb.w);
        float inter = fmaxf(ix2 - ix1, 0.0f) * fmaxf(iy2 - iy1, 0.0f);
        float uni = fmaxf(rarea + carea - inter, 1e-9f);
        bool hit = (inter > NMS_TH * uni) && (c > r) && (c < PPRE);
        bits |= (hit ? 1u : 0u) << t;
      }
      out[w] = bits;
    }
  }
}

// one wave32 per batch; suppression state held in 2 VGPRs per lane (2048 bits)
__global__ __launch_bounds__(32) void k_scan(const uint32_t* __restrict__ maskM,
                                             uint32_t* __restrict__ keep) {
  int b = blockIdx.x;
  int lane = threadIdx.x;
  uint32_t s0 = 0u, s1 = 0u;  // words 2*lane and 2*lane+1
  const uint32_t* mb = maskM + (size_t)b * PPRE * KWORDS;
  uint2 nxt = *(const uint2*)(mb + 2 * lane);
  for (int i = 0; i < PPRE; ++i) {
    uint2 cur = nxt;
    if (i + 1 < PPRE)
      nxt = *(const uint2*)(mb + (size_t)(i + 1) * KWORDS + 2 * lane);  // prefetch
    int w = i >> 5;
    uint32_t a0 = __shfl(s0, w >> 1);
    uint32_t a1 = __shfl(s1, w >> 1);
    uint32_t word = (w & 1) ? a1 : a0;
    uint32_t alive = ((word >> (i & 31)) & 1u) ^ 1u;
    if (lane == 0) keep[b * PPRE + i] = alive;
    if (alive) { s0 |= cur.x; s1 |= cur.y; }  // uniform branch across wave
  }
}

__global__ __launch_bounds__(256) void k_output(const float* __restrict__ tsc,
                                                const int* __restrict__ tidx,
                                                const float* __restrict__ tbox,
                                                const uint32_t* __restrict__ keep,
                                                float* __restrict__ out) {
  __shared__ int pos[PPRE];
  int b = blockIdx.x;
  if (threadIdx.x == 0) {
    int kc = 0;
    for (int i = 0; i < PPRE; ++i) kc += (keep[b * PPRE + i] != 0u);
    int kr = 0, ur = 0;
    for (int i = 0; i < PPRE; ++i) {
      if (keep[b * PPRE + i]) pos[i] = kr++;
      else                    pos[i] = kc + (ur++);
    }
  }
  __syncthreads();
  float* oScore = out;                      // [B][1000]
  float* oBox   = out + BATCH * PPOST;      // [B][1000][4]
  float* oIdx   = out + BATCH * PPOST * 5;  // [B][1000]
  float* oVal   = out + BATCH * PPOST * 6;  // [B][1000]
  for (int i = threadIdx.x; i < PPRE; i += 256) {
    int p = pos[i];
    if (p < PPOST) {
      int v = keep[b * PPRE + i] ? 1 : 0;
      float fv = (float)v;
      oScore[b * PPOST + p] = tsc[(size_t)b * PPRE + i] * fv;
      const float* bx = tbox + ((size_t)b * PPRE + i) * 4;
      float* ob = oBox + ((size_t)b * PPOST + p) * 4;
      ob[0] = bx[0] * fv; ob[1] = bx[1] * fv; ob[2] = bx[2] * fv; ob[3] = bx[3] * fv;
      oIdx[b * PPOST + p] = v ? (float)tidx[(size_t)b * PPRE + i] : -1.0f;
      oVal[b * PPOST + p] = fv;
    }
  }
}

extern "C" void kernel_launch(void* const* d_in, const int* in_sizes, int n_in,
                              void* d_out, int out_size, void* d_ws, size_t ws_size,
                              hipStream_t stream) {
  const float* bbox    = (const float*)d_in[0];  // (8,200000,4)
  const float* score   = (const float*)d_in[1];  // (8,200000)
  const float* anchors = (const float*)d_in[2];  // (200000,4)

  char* ws = (char*)d_ws;
  uint64_t* cand = (uint64_t*)(ws + WS_CAND);
  uint32_t* hist = (uint32_t*)(ws + WS_HIST);
  uint32_t* thrT = (uint32_t*)(ws + WS_THR);
  uint32_t* ccnt = (uint32_t*)(ws + WS_CCNT);
  float*    tsc  = (float*)(ws + WS_TSC);
  int*      tidx = (int*)(ws + WS_TIDX);
  float*    tbox = (float*)(ws + WS_TBOX);
  uint32_t* keep = (uint32_t*)(ws + WS_KEEP);
  uint32_t* maskM = (uint32_t*)(ws + WS_MASK);
  float* out = (float*)d_out;

  int chunks = (NPROP + 2047) / 2048;

  k_init<<<dim3((BATCH * NBUCKET + 255) / 256), 256, 0, stream>>>(hist, ccnt);
  k_hist<<<dim3(chunks, BATCH), 256, 0, stream>>>(score, hist);
  k_thresh<<<dim3(BATCH), 32, 0, stream>>>(hist, thrT);
  k_compact<<<dim3(chunks, BATCH), 256, 0, stream>>>(score, thrT, ccnt, cand);
  k_sort_decode<<<dim3(BATCH), 1024, 0, stream>>>(score, bbox, anchors, ccnt, cand,
                                                  tsc, tidx, tbox);
  k_mask<<<dim3((PPRE + MROWS - 1) / MROWS, BATCH), 256, 0, stream>>>(tbox, maskM);
  k_scan<<<dim3(BATCH), 32, 0, stream>>>(maskM, keep);
  k_output<<<dim3(BATCH), 256, 0, stream>>>(tsc, tidx, tbox, keep, out);
}